// DPPLayer_52286931862272
// MI455X (gfx1250) — compile-verified
//
#include <hip/hip_runtime.h>
#include <stdint.h>

// DPP sampling on gfx1250: CholeskyQR-based re-orthonormalization using
// V_WMMA_F32_16X16X4_F32 (full f32 precision, matching the reference QR) and
// async global->LDS copies (ASYNCcnt, GVS saddr addressing) for panel staging.
// Fixed 192-iteration graph-safe launch schedule with device-side guards.

constexpr int NN = 2048;   // n
constexpr int KM = 192;    // KMAX

typedef __attribute__((ext_vector_type(2))) float v2f;
typedef __attribute__((ext_vector_type(8))) float v8f;

struct DppState { int kk; int k; int item; int j; float pivot; int pad[3]; };

__device__ __forceinline__ uint32_t hash32(uint32_t x){
  x ^= x >> 16; x *= 0x7feb352du; x ^= x >> 15; x *= 0x846ca68bu; x ^= x >> 16; return x;
}
__device__ __forceinline__ float rng01(uint32_t seed, uint32_t i){
  uint32_t h = hash32(seed * 0x9E3779B9u ^ hash32(i + 0x85ebca6bu));
  return (float)(h >> 8) * (1.0f / 16777216.0f);
}

// Async global->LDS, GV mode (64-bit VGPR address). Tracked by ASYNCcnt.
__device__ __forceinline__ void async_g2l_b128(uint32_t lds_byte_off, const float* gsrc){
  asm volatile("global_load_async_to_lds_b128 %0, %1, off"
               :: "v"(lds_byte_off), "v"(gsrc) : "memory");
}
// Async global->LDS, GVS mode: global = SADDR + VOFF + IMM, lds = VDST + IMM.
// (IMM is added on BOTH sides per the ISA pseudocode, so callers pre-subtract
// it from the LDS offset.)
__device__ __forceinline__ void async_g2l_b32_s(uint32_t lds_byte_off, uint32_t goff,
                                                const float* sbase, int imm){
  asm volatile("global_load_async_to_lds_b32 %0, %1, %2 offset:%3"
               :: "v"(lds_byte_off), "v"(goff), "s"(sbase), "i"(imm) : "memory");
}
__device__ __forceinline__ void wait_async0(){
  asm volatile("s_wait_asynccnt 0x0" ::: "memory");
}

// ---------------- init: Bernoulli selection, zero output -------------------
__global__ void k_init(const float* __restrict__ e, const int* __restrict__ seedp,
                       int* __restrict__ flags, float* __restrict__ out){
  int i = blockIdx.x * blockDim.x + threadIdx.x;
  if (i < NN){
    float ei = e[i];
    flags[i] = rng01((uint32_t)seedp[0], (uint32_t)i) < (ei / (ei + 1.0f)) ? 1 : 0;
    out[i] = 0.0f;
  }
}

// ---------------- select: prefix-sum -> order[0..KM) -----------------------
__global__ void k_select(const int* __restrict__ flags, int* __restrict__ order,
                         DppState* __restrict__ st){
  __shared__ int S[NN];
  int t = threadIdx.x;                       // 1024 threads, 2 elems each
  S[t] = flags[t]; S[t + 1024] = flags[t + 1024];
  __syncthreads();
  for (int off = 1; off < NN; off <<= 1){
    int a0 = (t >= off)          ? S[t - off]          : 0;
    int a1 = (t + 1024 >= off)   ? S[t + 1024 - off]   : 0;
    __syncthreads();
    S[t] += a0; S[t + 1024] += a1;
    __syncthreads();
  }
  int k  = S[NN - 1];
  int kk = k < KM ? k : KM;
  for (int i = t; i < NN; i += 1024){
    int sel  = flags[i];
    int rank = sel ? (S[i] - 1) : (k + (i - S[i]));   // stable argsort(~index)
    if (rank < KM) order[rank] = i;
  }
  if (t == 0){ st->kk = kk; st->k = k; }
}

// ---------------- gather V0 = vecs[:, order] * colmask ---------------------
__global__ void k_gather(const float* __restrict__ vecs, const int* __restrict__ order,
                         const DppState* __restrict__ st, float* __restrict__ V){
  int idx = blockIdx.x * blockDim.x + threadIdx.x;
  if (idx >= NN * KM) return;
  int r = idx / KM, c = idx % KM;
  V[idx] = (c < st->kk) ? vecs[r * NN + order[c]] : 0.0f;
}

// ---------------- p[r] = sum_c V[r,c]^2  (one wave32 per row) --------------
__global__ void k_rownorm(const float* __restrict__ V, const DppState* __restrict__ st,
                          float* __restrict__ p, int it){
  if (it >= st->kk) return;
  int wave = threadIdx.x >> 5, lane = threadIdx.x & 31;
  int r = blockIdx.x * 8 + wave;
  const float* row = V + (size_t)r * KM;
  float s = 0.0f;
  for (int c = lane; c < KM; c += 32){ float x = row[c]; s += x * x; }
  for (int m = 16; m > 0; m >>= 1) s += __shfl_xor(s, m, 32);
  if (lane == 0) p[r] = s;
}

// ---------------- sample item, pivot column j, scale vector s --------------
__global__ void k_sample(const float* __restrict__ V, const float* __restrict__ p,
                         DppState* __restrict__ st, float* __restrict__ s,
                         float* __restrict__ out, const int* __restrict__ seedp, int it){
  if (it >= st->kk) return;
  __shared__ float S[NN];
  __shared__ int sh_item, sh_j;
  int t = threadIdx.x;
  S[t] = p[t]; S[t + 1024] = p[t + 1024];
  if (t == 0){ sh_item = NN - 1; sh_j = 1 << 30; }
  __syncthreads();
  for (int off = 1; off < NN; off <<= 1){
    float a0 = (t >= off)        ? S[t - off]        : 0.0f;
    float a1 = (t + 1024 >= off) ? S[t + 1024 - off] : 0.0f;
    __syncthreads();
    S[t] += a0; S[t + 1024] += a1;
    __syncthreads();
  }
  float total = S[NN - 1];
  float u   = rng01((uint32_t)seedp[0] ^ 0xdecafbadu, (uint32_t)it);
  float thr = u * total;
  if (S[t] > thr)        atomicMin(&sh_item, t);
  if (S[t + 1024] > thr) atomicMin(&sh_item, t + 1024);
  __syncthreads();
  int item = sh_item;
  int m = st->kk - it;
  if (t < KM){
    float av = fabsf(V[(size_t)item * KM + t]);
    if (av > 0.0f && t < m) atomicMin(&sh_j, t);
  }
  __syncthreads();
  int j = (sh_j < KM) ? sh_j : 0;
  float pivot = V[(size_t)item * KM + j];
  if (t < KM){
    int src = (t >= j) ? t + 1 : t; if (src > KM - 1) src = KM - 1;
    float mask = (t < m - 1) ? 1.0f : 0.0f;
    float vd = V[(size_t)item * KM + src] * mask;
    s[t] = (pivot != 0.0f) ? vd / pivot : 0.0f;
  }
  if (t == 0){
    out[item] = 1.0f;
    st->item = item; st->j = j; st->pivot = pivot;
  }
}

// ---------------- V2 = (shift-remove col j) - Vj * s, masked ---------------
__global__ void k_rank1(const float* __restrict__ V, const float* __restrict__ s,
                        const DppState* __restrict__ st, float* __restrict__ V2, int it){
  if (it >= st->kk) return;
  int idx = blockIdx.x * blockDim.x + threadIdx.x;
  if (idx >= NN * KM) return;
  int r = idx / KM, c = idx % KM;
  int m = st->kk - it, j = st->j;
  int src = (c >= j) ? c + 1 : c; if (src > KM - 1) src = KM - 1;
  float mask = (c < m - 1) ? 1.0f : 0.0f;
  float vd = V[(size_t)r * KM + src] * mask;
  V2[idx] = (vd - V[(size_t)r * KM + j] * s[c]) * mask;
}

// ---------------- G = V2^T * V2 via V_WMMA_F32_16X16X4_F32 -----------------
// f32 WMMA layouts (wave32):
//   A (16x4):  lane l, vgpr v -> M = l&15, K = v + 2*(l>>4)
//   B (4x16):  lane l, vgpr v -> K = v + 2*(l>>4), N = l&15
//   C/D:       lane l, vgpr v -> M = v + 8*(l>>4), N = l&15
// With LDS panel stored transposed VsT[col][K], both A and B fragments are a
// single contiguous 8-byte LDS load (compiler fuses to ds_load_2addr_b64).
constexpr int TSP = 34;    // padded K-stride (even -> 8B-aligned v2f loads)
constexpr int GCHUNK = 32; // K rows staged per chunk
__global__ void k_gram(const float* __restrict__ V2, const DppState* __restrict__ st,
                       float* __restrict__ G, int it){
  if (it >= st->kk) return;
  __shared__ float VsT[KM * TSP];            // [col][K] transposed panel, 26.1 KB
  int t = threadIdx.x;
  int wave = t >> 5, lane = t & 31;
  int tile = blockIdx.x * 8 + wave;          // 18 blocks * 8 waves = 144 tiles
  int mi = (tile / 12) * 16, mj = (tile % 12) * 16;
  int lh = lane >> 4, ll = lane & 15;
  uint32_t lbase = (uint32_t)(uintptr_t)&VsT[0];

  // Staging address plan (all k0-invariant): flat element u handled by this
  // thread is idx = t + 256u; global byte offset = idx*4 = t*4 + u*1024 with
  // u*1024 as the instruction immediate, t*4 as a single VGPR, and the chunk
  // base V2 + k0*KM in an SGPR pair (one scalar add per chunk). The LDS
  // transposed-scatter offset pre-subtracts the immediate (hardware adds it
  // back on the LDS side too).
  constexpr int NPT = (GCHUNK * KM) / 256;   // 24 elements per thread
  uint32_t loff2[NPT];
#pragma unroll
  for (int u = 0; u < NPT; ++u){
    int idx = t + u * 256;
    int rr = idx / KM, cc = idx - rr * KM;   // transposing scatter: [cc][rr]
    loff2[u] = lbase + (uint32_t)(cc * TSP + rr) * 4u - (uint32_t)(u * 1024);
  }
  uint32_t goff = (uint32_t)t * 4u;

  v8f acc = {};
  for (int k0 = 0; k0 < NN; k0 += GCHUNK){
    const float* base = V2 + (size_t)k0 * KM;   // uniform -> SGPR pair
    __syncthreads();
#pragma unroll
    for (int u = 0; u < NPT; ++u)
      async_g2l_b32_s(loff2[u], goff, base, u * 1024);
    if (k0 + GCHUNK < NN) __builtin_prefetch(base + GCHUNK * KM + t, 0, 1);
    wait_async0();
    __syncthreads();
#pragma unroll
    for (int s4 = 0; s4 < 8; ++s4){
      int kb = s4 * 4 + 2 * lh;
      v2f a = *(const v2f*)&VsT[(mi + ll) * TSP + kb];   // A = V2^T tile
      v2f b = *(const v2f*)&VsT[(mj + ll) * TSP + kb];   // B = V2 tile
      acc = __builtin_amdgcn_wmma_f32_16x16x4_f32(false, a, false, b,
                                                  (short)0, acc, false, false);
    }
  }
#pragma unroll
  for (int v = 0; v < 8; ++v)
    G[(size_t)(mi + v + 8 * lh) * KM + mj + ll] = acc[v];
}

// ---------------- Cholesky (upper R, GS sign convention) + R^-T ------------
// Produces RT with RT[n][k] = (R^-1)[k][n]  (transposed for vectorized B
// fragment loads in k_apply; back-substitution writes contiguous rows).
__global__ void k_chol(float* __restrict__ G, float* __restrict__ RT,
                       const DppState* __restrict__ st, int it){
  if (it >= st->kk) return;
  int t = threadIdx.x;
  int m = st->kk - it;
  __shared__ float dinv;
  for (int c = t; c < KM; c += 256){
    if (c >= m - 1) G[(size_t)c * KM + c] = 1.0f;     // identity-extend masked cols
    else            G[(size_t)c * KM + c] += 1e-12f;  // regularize
  }
  __syncthreads();
  for (int kd = 0; kd < KM; ++kd){
    if (t == 0){
      float d = sqrtf(fmaxf(G[(size_t)kd * KM + kd], 1e-20f));
      G[(size_t)kd * KM + kd] = d;
      dinv = 1.0f / d;
    }
    __syncthreads();
    for (int jj = kd + 1 + t; jj < KM; jj += 256) G[(size_t)kd * KM + jj] *= dinv;
    __syncthreads();
    for (int idx = t; idx < KM * KM; idx += 256){
      int i = idx / KM, jj = idx % KM;
      if (i > kd && jj >= i)
        G[(size_t)i * KM + jj] -= G[(size_t)kd * KM + i] * G[(size_t)kd * KM + jj];
    }
    __syncthreads();
  }
  // Back-substitution: one column of R^-1 (= one contiguous row of RT) per
  // thread.
  for (int jcol = t; jcol < KM; jcol += 256){
    float* RTrow = RT + (size_t)jcol * KM;
    for (int i = jcol + 1; i < KM; ++i) RTrow[i] = 0.0f;
    RTrow[jcol] = 1.0f / G[(size_t)jcol * KM + jcol];
    for (int i = jcol - 1; i >= 0; --i){
      float sum = 0.0f;
      for (int tt = i + 1; tt <= jcol; ++tt)
        sum += G[(size_t)i * KM + tt] * RTrow[tt];
      RTrow[i] = -sum / G[(size_t)i * KM + i];
    }
  }
}

// ---------------- V = V2 * R^-1 via V_WMMA_F32_16X16X4_F32 -----------------
constexpr int ASP = 196;   // padded row stride for A panel (even)
__global__ void k_apply(const float* __restrict__ V2, const float* __restrict__ RT,
                        const DppState* __restrict__ st, float* __restrict__ V, int it){
  if (it >= st->kk) return;
  __shared__ float As[16 * ASP];             // 12.5 KB: rows mi..mi+15 of V2
  int t = threadIdx.x;                       // 384 threads = 12 waves
  int wave = t >> 5, lane = t & 31;
  int mi = blockIdx.x * 16;                  // 128 blocks cover M=2048
  int nj = wave * 16;                        // 12 waves cover N=192
  int lh = lane >> 4, ll = lane & 15;
  uint32_t lbase = (uint32_t)(uintptr_t)&As[0];
  // Async b128 stage (layout-preserving): 16x192 floats = 768 float4s.
  for (int idx = t; idx < 768; idx += 384){
    int rr = idx / 48, c4 = idx - rr * 48;
    async_g2l_b128(lbase + (uint32_t)(rr * ASP + c4 * 4) * 4u,
                   &V2[(size_t)(mi + rr) * KM + c4 * 4]);
  }
  wait_async0();
  __syncthreads();
  const float* Brow = RT + (size_t)(nj + ll) * KM;   // B[k][n]=RT[n][k]
  v8f acc = {};
#pragma unroll 4
  for (int kb = 0; kb < KM; kb += 4){
    int kr = kb + 2 * lh;
    v2f a = *(const v2f*)&As[ll * ASP + kr];
    v2f b = *(const v2f*)&Brow[kr];          // single 8B global load
    acc = __builtin_amdgcn_wmma_f32_16x16x4_f32(false, a, false, b,
                                                (short)0, acc, false, false);
  }
#pragma unroll
  for (int v = 0; v < 8; ++v)
    V[(size_t)(mi + v + 8 * lh) * KM + nj + ll] = acc[v];
}

// ---------------- finalize: k == n -> all ones -----------------------------
__global__ void k_final(const DppState* __restrict__ st, float* __restrict__ out){
  int i = blockIdx.x * blockDim.x + threadIdx.x;
  if (i < NN && st->k == NN) out[i] = 1.0f;
}

extern "C" void kernel_launch(void* const* d_in, const int* in_sizes, int n_in,
                              void* d_out, int out_size, void* d_ws, size_t ws_size,
                              hipStream_t stream){
  (void)in_sizes; (void)n_in; (void)out_size; (void)ws_size;
  const float* e     = (const float*)d_in[0];
  const float* vecs  = (const float*)d_in[1];
  const int*   seedp = (const int*)d_in[2];
  float* out = (float*)d_out;

  char* ws = (char*)d_ws;
  size_t off = 0;
  auto carve = [&](size_t bytes) -> void* {
    void* p = ws + off;
    off = (off + bytes + 255) & ~(size_t)255;
    return p;
  };
  DppState* st = (DppState*)carve(sizeof(DppState));
  int*   flags = (int*)  carve((size_t)NN * sizeof(int));
  int*   order = (int*)  carve((size_t)KM * sizeof(int));
  float* V     = (float*)carve((size_t)NN * KM * sizeof(float));
  float* V2    = (float*)carve((size_t)NN * KM * sizeof(float));
  float* G     = (float*)carve((size_t)KM * KM * sizeof(float));
  float* RT    = (float*)carve((size_t)KM * KM * sizeof(float));
  float* p     = (float*)carve((size_t)NN * sizeof(float));
  float* s     = (float*)carve((size_t)KM * sizeof(float));

  k_init  <<<NN / 256, 256, 0, stream>>>(e, seedp, flags, out);
  k_select<<<1, 1024, 0, stream>>>(flags, order, st);
  k_gather<<<(NN * KM) / 256, 256, 0, stream>>>(vecs, order, st, V);

  for (int it = 0; it < KM; ++it){
    k_rownorm<<<NN / 8, 256, 0, stream>>>(V, st, p, it);
    k_sample <<<1, 1024, 0, stream>>>(V, p, st, s, out, seedp, it);
    k_rank1  <<<(NN * KM) / 256, 256, 0, stream>>>(V, s, st, V2, it);
    k_gram   <<<18, 256, 0, stream>>>(V2, st, G, it);
    k_chol   <<<1, 256, 0, stream>>>(G, RT, st, it);
    k_apply  <<<NN / 16, 384, 0, stream>>>(V2, RT, st, V, it);
  }
  k_final<<<NN / 256, 256, 0, stream>>>(st, out);
}